// AttModel_75591424410289
// MI455X (gfx1250) — compile-verified
//
#include <hip/hip_runtime.h>
#include <hip/hip_bf16.h>
#include <math.h>
#include <cstdint>

// ---------------------------------------------------------------------------
// Model dims (from reference)
// ---------------------------------------------------------------------------
constexpr int BS      = 256;
constexpr int IN_F    = 66;
constexpr int D_MODEL = 512;
constexpr int KS      = 10;
constexpr int DCT_N   = 10;
constexpr int N_LAYER = 6;
constexpr int N_STAGE = 3;
constexpr int INPUT_N = 50;
constexpr int OUTPUT_N = 10;
constexpr int VL = KS + OUTPUT_N;               // 20
constexpr int VN = INPUT_N - KS - OUTPUT_N + 1; // 31
constexpr int MROWS = BS * IN_F;                // 16896 (divisible by 64)

typedef __attribute__((ext_vector_type(2))) float v2f;
typedef __attribute__((ext_vector_type(8))) float v8f;
typedef __attribute__((ext_vector_type(4))) unsigned int u32x4;
typedef __attribute__((ext_vector_type(8))) int i32x8;
typedef __attribute__((ext_vector_type(4))) int i32x4;

#define GLOBAL_AS __attribute__((address_space(1)))
#define LDS_AS    __attribute__((address_space(3)))

// fp32 WMMA: D(16x16) = A(16x4) * B(4x16) + C.  wave32, 8-arg form.
static __device__ inline v8f wmma_f32(v2f a, v2f b, v8f c) {
  return __builtin_amdgcn_wmma_f32_16x16x4_f32(false, a, false, b, (short)0, c,
                                               false, false);
}

#if __has_builtin(__builtin_amdgcn_tensor_load_to_lds)
#define USE_TDM 1
#elif __has_builtin(__builtin_amdgcn_global_load_async_to_lds_b128)
#define USE_ASYNC 1
#endif

// ---------------------------------------------------------------------------
// DCT matrix (orthonormal DCT-II; inverse == transpose)
// ---------------------------------------------------------------------------
__global__ void build_dct_kernel(float* dct) {
  int i = threadIdx.x;
  if (i < VL * VL) {
    int k = i / VL, c = i % VL;
    float w = (k == 0) ? sqrtf(1.0f / VL) : sqrtf(2.0f / VL);
    dct[i] = w * cosf(3.14159265358979323846f * (c + 0.5f) * k / (float)VL);
  }
}

// src_value[b, v, f*10+d] = sum_l dct[d,l] * src[b, v+l, f]
__global__ void src_value_kernel(const float* __restrict__ src,
                                 const float* __restrict__ dct,
                                 float* __restrict__ sv) {
  int idx = blockIdx.x * blockDim.x + threadIdx.x;
  if (idx >= BS * VN * IN_F * DCT_N) return;
  int fd = idx % (IN_F * DCT_N);
  int bv = idx / (IN_F * DCT_N);
  int v = bv % VN, b = bv / VN;
  int d = fd % DCT_N, f = fd / DCT_N;
  float s = 0.f;
  for (int l = 0; l < VL; ++l)
    s += dct[d * VL + l] * src[(b * INPUT_N + v + l) * IN_F + f];
  sv[idx] = s;
}

// dct_in[b, f*10+d] = sum_l dct[d,l]*src[b, idx2[l], f]
__global__ void dct_in_init_kernel(const float* __restrict__ src,
                                   const float* __restrict__ dct,
                                   float* __restrict__ di) {
  int idx = blockIdx.x * blockDim.x + threadIdx.x;
  if (idx >= BS * IN_F * DCT_N) return;
  int fd = idx % (IN_F * DCT_N);
  int b  = idx / (IN_F * DCT_N);
  int d = fd % DCT_N, f = fd / DCT_N;
  float s = 0.f;
  for (int l = 0; l < VL; ++l) {
    int t = (l < KS) ? (INPUT_N - KS + l) : (INPUT_N - 1);
    s += dct[d * VL + l] * src[(b * INPUT_N + t) * IN_F + f];
  }
  di[idx] = s;
}

// ---------------------------------------------------------------------------
// Implicit-GEMM conv1d ('NCH'/'OIH', VALID) + ReLU.  One 16x16 tile per wave.
// KW is a template constant so the %KW / /KW in the K loop strength-reduce.
// ---------------------------------------------------------------------------
template <int KW>
__global__ void conv_gemm_kernel(const float* __restrict__ inp,
                                 const float* __restrict__ w,
                                 float* __restrict__ out,
                                 int IC, int Tin, int t0, int Tout,
                                 int Kdim, float scale) {
  const int lane = threadIdx.x, wv = threadIdx.y;
  const int tm = (blockIdx.x * 4 + wv) * 16;
  const int tn = blockIdx.y * 16;
  const int m = tm + (lane & 15);
  const int n = tn + (lane & 15);
  const int kh = (lane >> 4) * 2;
  const int b = m / Tout, t = m % Tout;
  const float* irow = inp + (size_t)(b * Tin + t0 + t) * IC;
  v8f acc = {};
  for (int k0 = 0; k0 < Kdim; k0 += 4) {
    int kk0 = k0 + kh, kk1 = kk0 + 1;
    v2f a, bb;
    a.x = irow[(kk0 % KW) * IC + (kk0 / KW)] * scale;
    a.y = irow[(kk1 % KW) * IC + (kk1 / KW)] * scale;
    bb.x = w[(size_t)n * Kdim + kk0];
    bb.y = w[(size_t)n * Kdim + kk1];
    acc = wmma_f32(a, bb, acc);
  }
  const int ms = tm + ((lane >> 4) << 3);
  for (int r = 0; r < 8; ++r) {
    float v = acc[r];
    out[(size_t)(ms + r) * D_MODEL + n] = v > 0.f ? v : 0.f;
  }
}

// score/att: one block (32 threads) per batch
__global__ void score_att_kernel(const float* __restrict__ q2,
                                 const float* __restrict__ key2,
                                 float* __restrict__ att) {
  __shared__ float s[32];
  int b = blockIdx.x, k = threadIdx.x;
  float sc = 0.f;
  if (k < VN) {
    const float* q = q2 + (size_t)b * D_MODEL;
    const float* kk = key2 + (size_t)(b * VN + k) * D_MODEL;
    for (int d = 0; d < D_MODEL; ++d) sc += q[d] * kk[d];
    sc += 1e-15f;
  }
  s[k] = (k < VN) ? sc : 0.f;
  __syncthreads();
  float tot = 0.f;
  for (int j = 0; j < VN; ++j) tot += s[j];
  if (k < VN) att[b * VN + k] = sc / tot;
}

// dct_att[b, fd] = sum_k att[b,k]*sv[b,k,fd]
__global__ void dct_att_kernel(const float* __restrict__ att,
                               const float* __restrict__ sv,
                               float* __restrict__ da) {
  int idx = blockIdx.x * blockDim.x + threadIdx.x;
  if (idx >= BS * IN_F * DCT_N) return;
  int fd = idx % (IN_F * DCT_N);
  int b  = idx / (IN_F * DCT_N);
  float s = 0.f;
  for (int k = 0; k < VN; ++k)
    s += att[b * VN + k] * sv[(size_t)(b * VN + k) * (IN_F * DCT_N) + fd];
  da[idx] = s;
}

// x[(b*66+f)*20+d] = d<10 ? dct_in : dct_att
__global__ void build_x_kernel(const float* __restrict__ di,
                               const float* __restrict__ da,
                               float* __restrict__ x) {
  int idx = blockIdx.x * blockDim.x + threadIdx.x;
  if (idx >= MROWS * VL) return;
  int d = idx % VL, bf = idx / VL;
  x[idx] = (d < DCT_N) ? di[bf * DCT_N + d] : da[bf * DCT_N + (d - DCT_N)];
}

// h0 = tanh(x @ Win) (+ his).  M=16896,K=20,N=512; all exact multiples.
__global__ void h0_kernel(const float* __restrict__ x,
                          const float* __restrict__ Win,
                          const float* __restrict__ his,
                          float* __restrict__ h0) {
  const int lane = threadIdx.x, wv = threadIdx.y;
  const int tm = (blockIdx.x * 4 + wv) * 16;
  const int tn = blockIdx.y * 16;
  const int m = tm + (lane & 15);
  const int n = tn + (lane & 15);
  const int kh = (lane >> 4) * 2;
  v8f acc = {};
  for (int k0 = 0; k0 < VL; k0 += 4) {
    int kk = k0 + kh;
    v2f a, b;
    a.x = x[(size_t)m * VL + kk];
    a.y = x[(size_t)m * VL + kk + 1];
    b.x = Win[(size_t)kk * D_MODEL + n];
    b.y = Win[(size_t)(kk + 1) * D_MODEL + n];
    acc = wmma_f32(a, b, acc);
  }
  const int ms = tm + ((lane >> 4) << 3);
  for (int r = 0; r < 8; ++r) {
    float v = tanhf(acc[r]);
    if (his) v += his[(size_t)(ms + r) * D_MODEL + n];
    h0[(size_t)(ms + r) * D_MODEL + n] = v;
  }
}

// m[b] = 0.9 * (A @ h[b]) + 0.1 * h0[b].  Per-batch 66x66 @ 66x512, K pad 68.
__global__ void propagate_kernel(const float* __restrict__ Ast,
                                 const float* __restrict__ hin,
                                 const float* __restrict__ h0,
                                 float* __restrict__ mout) {
  const int lane = threadIdx.x, wv = threadIdx.y;
  const int b = blockIdx.z;
  const int tm = (blockIdx.x * 4 + wv) * 16;
  if (tm >= 80) return;  // wave-uniform exit; EXEC stays all-ones elsewhere
  const int tn = blockIdx.y * 16;
  const int row = tm + (lane & 15);
  const int n = tn + (lane & 15);
  const int kh = (lane >> 4) * 2;
  const float* hb = hin + (size_t)b * IN_F * D_MODEL;
  v8f acc = {};
  for (int k0 = 0; k0 < 68; k0 += 4) {
    int kA = k0 + kh, kB = kA + 1;
    v2f a, bb;
    a.x = (row < IN_F && kA < IN_F) ? Ast[row * IN_F + kA] : 0.f;
    a.y = (row < IN_F && kB < IN_F) ? Ast[row * IN_F + kB] : 0.f;
    bb.x = (kA < IN_F) ? hb[(size_t)kA * D_MODEL + n] : 0.f;
    bb.y = (kB < IN_F) ? hb[(size_t)kB * D_MODEL + n] : 0.f;
    acc = wmma_f32(a, bb, acc);
  }
  const int ms = tm + ((lane >> 4) << 3);
  for (int r = 0; r < 8; ++r) {
    int node = ms + r;
    if (node < IN_F) {
      size_t o = ((size_t)b * IN_F + node) * D_MODEL + n;
      mout[o] = 0.9f * acc[r] + 0.1f * h0[o];
    }
  }
}

// ---------------------------------------------------------------------------
// h = tanh((1-beta)*m + beta*(m @ W)); 16896x512x512.
// Wave computes a 16(M)x64(N) strip (4 accumulators -> 4x A-register reuse).
// W chunk (KC=32 rows x 64 cols) staged into LDS via the Tensor Data Mover
// (one descriptor per block chunk, S_WAIT_TENSORCNT), falling back to
// GLOBAL_LOAD_ASYNC_TO_LDS_B128 (+S_WAIT_ASYNCCNT), then to plain copies.
// ---------------------------------------------------------------------------
__global__ void layer_kernel(const float* __restrict__ mbuf,
                             const float* __restrict__ W,
                             float* __restrict__ hout, float beta) {
  constexpr int KC = 32;
  constexpr int NT = 64;
  __shared__ float Bs[KC * NT];  // 8 KB
  const int lane = threadIdx.x, wv = threadIdx.y;
  const int tid = wv * 32 + lane;
  const int tm = (blockIdx.x * 4 + wv) * 16;
  const int tn0 = blockIdx.y * NT;
  const int m = tm + (lane & 15);
  const int nn = lane & 15;
  const int kh = (lane >> 4) * 2;
  const float* arow = mbuf + (size_t)m * D_MODEL;
  v8f acc[4] = {};
  (void)tid;
  for (int kc = 0; kc < D_MODEL; kc += KC) {
#if defined(USE_TDM)
    if (wv == 0) {
      // D# group0: count=1 | lds_addr | global_addr | type=2
      unsigned ldsoff = (unsigned)(uintptr_t)(LDS_AS float*)Bs;
      unsigned long long va =
          (unsigned long long)(uintptr_t)(W + (size_t)kc * D_MODEL + tn0);
      u32x4 g0;
      g0.x = 1u;                                   // count=1 (valid), user mode
      g0.y = ldsoff;                               // lds_addr (bytes)
      g0.z = (unsigned)(va & 0xffffffffu);         // global_addr[31:0]
      g0.w = (unsigned)((va >> 32) & 0x1ffffffu) | (2u << 30);  // [56:32]|type=2
      // D# group1: data_size=4B; tensor 512x512; tile 64(X) x 32(Y); stride 512
      i32x8 g1;
      g1[0] = (int)(2u << 16);        // workgroup_mask=0, data_size=2 (4B)
      g1[1] = (int)(512u << 16);      // tensor_dim0[15:0] << 16
      g1[2] = (int)(512u << 16);      // tensor_dim0[31:16]=0 | tensor_dim1[15:0]<<16
      g1[3] = (int)(64u << 16);       // tensor_dim1[31:16]=0 | tile_dim0=64
      g1[4] = 32;                     // tile_dim1=32, tile_dim2=0
      g1[5] = 512;                    // tensor_dim0_stride[31:0]
      g1[6] = 0;                      // stride0[47:32]=0, stride1[15:0]=0
      g1[7] = 0;
      i32x4 z4 = {0, 0, 0, 0};
      i32x8 z8 = {0, 0, 0, 0, 0, 0, 0, 0};
      __builtin_amdgcn_tensor_load_to_lds(g0, g1, z4, z4, z8, 0);
#if __has_builtin(__builtin_amdgcn_s_wait_tensorcnt)
      __builtin_amdgcn_s_wait_tensorcnt(0);
#else
      asm volatile("s_wait_tensorcnt 0x0" ::: "memory");
#endif
    }
#elif defined(USE_ASYNC)
#pragma unroll
    for (int i = 0; i < 4; ++i) {
      int q = tid + 128 * i;          // float4 index within 32x64 tile
      int kr = q >> 4;
      int nc = (q & 15) << 2;
      __builtin_amdgcn_global_load_async_to_lds_b128(
          (GLOBAL_AS void*)(uintptr_t)(W + (size_t)(kc + kr) * D_MODEL + tn0 + nc),
          (LDS_AS void*)((LDS_AS float*)Bs + kr * NT + nc), 0, 0);
    }
#if __has_builtin(__builtin_amdgcn_s_wait_asynccnt)
    __builtin_amdgcn_s_wait_asynccnt(0);
#else
    asm volatile("s_wait_asynccnt 0x0" ::: "memory");
#endif
#else
    for (int q = tid; q < KC * NT / 4; q += 128) {
      int kr = q >> 4;
      int nc = (q & 15) << 2;
      *(float4*)&Bs[kr * NT + nc] =
          *(const float4*)&W[(size_t)(kc + kr) * D_MODEL + tn0 + nc];
    }
#endif
    __syncthreads();
#pragma unroll
    for (int k0 = 0; k0 < KC; k0 += 4) {
      int kk = kc + k0 + kh;
      v2f a;
      a.x = arow[kk];
      a.y = arow[kk + 1];
      const int rb = (k0 + kh) * NT + nn;
#pragma unroll
      for (int j = 0; j < 4; ++j) {
        v2f b;
        b.x = Bs[rb + 16 * j];
        b.y = Bs[rb + NT + 16 * j];
        acc[j] = wmma_f32(a, b, acc[j]);
      }
    }
    __syncthreads();
  }
  const int ms = tm + ((lane >> 4) << 3);
  const float omb = 1.0f - beta;
#pragma unroll
  for (int j = 0; j < 4; ++j) {
    const int n = tn0 + 16 * j + nn;
#pragma unroll
    for (int r = 0; r < 8; ++r) {
      size_t o = (size_t)(ms + r) * D_MODEL + n;
      hout[o] = tanhf(omb * mbuf[o] + beta * acc[j][r]);
    }
  }
}

// outb = x + h @ Wout.  N=20 (2 tiles, masked).
__global__ void wout_kernel(const float* __restrict__ x,
                            const float* __restrict__ h,
                            const float* __restrict__ Wout,
                            float* __restrict__ outb) {
  const int lane = threadIdx.x, wv = threadIdx.y;
  const int tm = (blockIdx.x * 4 + wv) * 16;
  const int tn = blockIdx.y * 16;
  const int m = tm + (lane & 15);
  const int n = tn + (lane & 15);
  const int kh = (lane >> 4) * 2;
  const bool nv = n < VL;
  v8f acc = {};
  for (int k0 = 0; k0 < D_MODEL; k0 += 4) {
    int kk = k0 + kh;
    v2f a, b;
    a.x = h[(size_t)m * D_MODEL + kk];
    a.y = h[(size_t)m * D_MODEL + kk + 1];
    b.x = nv ? Wout[kk * VL + n] : 0.f;
    b.y = nv ? Wout[(kk + 1) * VL + n] : 0.f;
    acc = wmma_f32(a, b, acc);
  }
  const int ms = tm + ((lane >> 4) << 3);
  if (nv)
    for (int r = 0; r < 8; ++r) {
      int mm = ms + r;
      outb[(size_t)mm * VL + n] = x[(size_t)mm * VL + n] + acc[r];
    }
}

// out_gcn[b,l,f] = sum_{d<10} dct[d,l] * outb[(b*66+f)*20+d]   (idct = dct^T)
__global__ void finalize_kernel(const float* __restrict__ outb,
                                const float* __restrict__ dct,
                                float* __restrict__ dst) {
  int idx = blockIdx.x * blockDim.x + threadIdx.x;
  if (idx >= BS * VL * IN_F) return;
  int f = idx % IN_F;
  int bl = idx / IN_F;
  int l = bl % VL, b = bl / VL;
  float s = 0.f;
  for (int d = 0; d < DCT_N; ++d)
    s += dct[d * VL + l] * outb[((size_t)b * IN_F + f) * VL + d];
  dst[idx] = s;
}

// dct_in(next) = outb[:, :, :10]
__global__ void update_dct_in_kernel(const float* __restrict__ outb,
                                     float* __restrict__ di) {
  int idx = blockIdx.x * blockDim.x + threadIdx.x;
  if (idx >= BS * IN_F * DCT_N) return;
  int d = idx % DCT_N, bf = idx / DCT_N;
  di[idx] = outb[(size_t)bf * VL + d];
}

// ---------------------------------------------------------------------------
extern "C" void kernel_launch(void* const* d_in, const int* in_sizes, int n_in,
                              void* d_out, int out_size, void* d_ws,
                              size_t ws_size, hipStream_t stream) {
  (void)in_sizes; (void)n_in; (void)out_size; (void)ws_size;
  const float* src   = (const float*)d_in[0];
  const float* cQw1  = (const float*)d_in[1];
  const float* cQw2  = (const float*)d_in[2];
  const float* cKw1  = (const float*)d_in[3];
  const float* cKw2  = (const float*)d_in[4];
  const float* gWin  = (const float*)d_in[5];
  const float* gA    = (const float*)d_in[6];
  const float* gWl   = (const float*)d_in[7];
  const float* gWout = (const float*)d_in[8];
  float* out = (float*)d_out;

  // Workspace layout (floats)
  float* ws = (float*)d_ws;
  size_t off = 0;
  float* dct  = ws + off; off += VL * VL;
  float* sv   = ws + off; off += (size_t)BS * VN * IN_F * DCT_N;
  float* key1 = ws + off; off += (size_t)BS * 35 * D_MODEL;
  float* key2 = ws + off; off += (size_t)BS * VN * D_MODEL;
  float* q1   = ws + off; off += (size_t)BS * 5 * D_MODEL;
  float* q2   = ws + off; off += (size_t)BS * D_MODEL;
  float* att  = ws + off; off += (size_t)BS * VN;
  float* da   = ws + off; off += (size_t)BS * IN_F * DCT_N;
  float* di   = ws + off; off += (size_t)BS * IN_F * DCT_N;
  float* x    = ws + off; off += (size_t)MROWS * VL;
  float* h0b  = ws + off; off += (size_t)MROWS * D_MODEL;
  float* hb   = ws + off; off += (size_t)MROWS * D_MODEL;
  float* mb   = ws + off; off += (size_t)MROWS * D_MODEL;
  float* outb = ws + off; off += (size_t)MROWS * VL;

  const dim3 wblk(32, 4);  // 4 waves per block

  build_dct_kernel<<<1, 512, 0, stream>>>(dct);

  {
    int n = BS * VN * IN_F * DCT_N;
    src_value_kernel<<<(n + 255) / 256, 256, 0, stream>>>(src, dct, sv);
  }

  // key convs:  (256,66,40)->(256,512,35)->(256,512,31)
  conv_gemm_kernel<6><<<dim3(BS * 35 / 64, D_MODEL / 16), wblk, 0, stream>>>(
      src, cKw1, key1, IN_F, INPUT_N, 0, 35, IN_F * 6, 1e-3f);
  conv_gemm_kernel<5><<<dim3(BS * VN / 64, D_MODEL / 16), wblk, 0, stream>>>(
      key1, cKw2, key2, D_MODEL, 35, 0, VN, D_MODEL * 5, 1.0f);
  // query convs: (256,66,10)->(256,512,5)->(256,512,1)
  conv_gemm_kernel<6><<<dim3(BS * 5 / 64, D_MODEL / 16), wblk, 0, stream>>>(
      src, cQw1, q1, IN_F, INPUT_N, 40, 5, IN_F * 6, 1e-3f);
  conv_gemm_kernel<5><<<dim3(BS / 64, D_MODEL / 16), wblk, 0, stream>>>(
      q1, cQw2, q2, D_MODEL, 5, 0, 1, D_MODEL * 5, 1.0f);

  score_att_kernel<<<BS, 32, 0, stream>>>(q2, key2, att);
  {
    int n = BS * IN_F * DCT_N;
    dct_att_kernel<<<(n + 255) / 256, 256, 0, stream>>>(att, sv, da);
    dct_in_init_kernel<<<(n + 255) / 256, 256, 0, stream>>>(src, dct, di);
  }

  for (int stage = 0; stage < N_STAGE; ++stage) {
    {
      int n = MROWS * VL;
      build_x_kernel<<<(n + 255) / 256, 256, 0, stream>>>(di, da, x);
    }
    const float* Win  = gWin + (size_t)stage * VL * D_MODEL;
    const float* Ast  = gA + (size_t)stage * IN_F * IN_F;
    const float* Wout = gWout + (size_t)stage * D_MODEL * VL;
    const float* his  = (stage == 0) ? nullptr : hb;

    h0_kernel<<<dim3(MROWS / 64, D_MODEL / 16), wblk, 0, stream>>>(x, Win, his,
                                                                   h0b);
    for (int l = 0; l < N_LAYER; ++l) {
      float beta = logf(0.5f / (float)(l + 1) + 1.0f);
      const float* W =
          gWl + ((size_t)stage * N_LAYER + l) * D_MODEL * D_MODEL;
      propagate_kernel<<<dim3(2, D_MODEL / 16, BS), wblk, 0, stream>>>(
          Ast, (l == 0) ? h0b : hb, h0b, mb);
      layer_kernel<<<dim3(MROWS / 64, D_MODEL / 64), wblk, 0, stream>>>(
          mb, W, hb, beta);
    }
    wout_kernel<<<dim3(MROWS / 64, 2), wblk, 0, stream>>>(x, hb, Wout, outb);
    {
      int n = BS * VL * IN_F;
      finalize_kernel<<<(n + 255) / 256, 256, 0, stream>>>(
          outb, dct, out + (size_t)stage * BS * VL * IN_F);
      int n2 = BS * IN_F * DCT_N;
      update_dct_in_kernel<<<(n2 + 255) / 256, 256, 0, stream>>>(outb, di);
    }
  }
}